// VertexShader_90872918048901
// MI455X (gfx1250) — compile-verified
//
#include <hip/hip_runtime.h>
#include <math.h>

// ---------------------------------------------------------------------------
// Mesh rasterizer for MI455X (gfx1250, wave32).
// Edge functions w_e(px,py) = A_e*px + B_e*py + C_e are evaluated for
// 16-pixel x 16-face blocks with V_WMMA_F32_16X16X4_F32:
//   A (16x4) = per-pixel [px, py, 1, 0],  B (4x16) = per-face [A_e;B_e;C_e;0]
// ---------------------------------------------------------------------------

typedef float v2f __attribute__((ext_vector_type(2)));
typedef float v8f __attribute__((ext_vector_type(8)));

#define RAST_EPS 1e-8f
#define CH 128                 // faces staged per LDS chunk (128*16 floats = 8KB)
#define RASTER_WAVES 8
#define RASTER_BLOCK (RASTER_WAVES * 32)

// --------------------------- kernel 1: vertex transform --------------------
__global__ void vertex_xform(const float* __restrict__ pos,
                             const float* __restrict__ Kmat,
                             const float* __restrict__ RT,
                             const int* __restrict__ pH, const int* __restrict__ pW,
                             float* __restrict__ verts, int V) {
  const int b = blockIdx.y;
  const int v = blockIdx.x * blockDim.x + threadIdx.x;
  if (v >= V) return;
  const float H = (float)*pH, W = (float)*pW;
  const float scale = 0.5f * fminf(H, W);
  const float* Kb = Kmat + (size_t)b * 9;
  const float* Rb = RT + (size_t)b * 12;
  const float fx = Kb[0] / scale, fy = Kb[4] / scale;
  const float p0x = -(Kb[2] - 0.5f * W) / scale;
  const float p0y = -(Kb[5] - 0.5f * H) / scale;
  const float* p = pos + ((size_t)b * V + v) * 3;
  const float X = p[0], Y = p[1], Z = p[2];
  // v_view[j] = sign[j] * (R[j,:]·pos + t[j]), sign = (-1,-1,1)
  const float vx = -(Rb[0] * X + Rb[1] * Y + Rb[2]  * Z + Rb[3]);
  const float vy = -(Rb[4] * X + Rb[5] * Y + Rb[6]  * Z + Rb[7]);
  const float vz =  (Rb[8] * X + Rb[9] * Y + Rb[10] * Z + Rb[11]);
  const float nx = fx * vx / vz + p0x;
  const float ny = fy * vy / vz + p0y;
  float* o = verts + ((size_t)b * V + v) * 4;
  o[0] = nx; o[1] = ny; o[2] = vz; o[3] = 0.0f;
}

// --------------------------- kernel 2: per-face setup ----------------------
// record layout (16 floats):
//  [0..2] edge0 A,B,C  [3..5] edge1  [6..8] edge2   (all premultiplied by 1/area)
//  [9..11] z0,z1,z2   [12..14] z1*z2, z0*z2, z0*z1   [15] face_ok flag
__global__ void face_setup(const int* __restrict__ faces,
                           const float* __restrict__ verts,
                           float* __restrict__ frec, int V, int Fn) {
  const int b = blockIdx.y;
  const int f = blockIdx.x * blockDim.x + threadIdx.x;
  if (f >= Fn) return;
  const int* fv = faces + ((size_t)b * Fn + f) * 3;
  const float* vb = verts + (size_t)b * V * 4;
  const float* v0 = vb + (size_t)fv[0] * 4;
  const float* v1 = vb + (size_t)fv[1] * 4;
  const float* v2 = vb + (size_t)fv[2] * 4;
  const float x0 = v0[0], y0 = v0[1], z0 = v0[2];
  const float x1 = v1[0], y1 = v1[1], z1 = v1[2];
  const float x2 = v2[0], y2 = v2[1], z2 = v2[2];
  const float area = (x1 - x0) * (y2 - y0) - (y1 - y0) * (x2 - x0);
  const bool ok = (area > RAST_EPS) && (z0 > 0.0f) && (z1 > 0.0f) && (z2 > 0.0f);
  const float ia = ok ? 1.0f / area : 0.0f;
  float* r = frec + ((size_t)b * Fn + f) * 16;
  r[0] = (y1 - y2) * ia; r[1] = (x2 - x1) * ia; r[2] = (x1 * y2 - y1 * x2) * ia;
  r[3] = (y2 - y0) * ia; r[4] = (x0 - x2) * ia; r[5] = (x2 * y0 - y2 * x0) * ia;
  r[6] = (y0 - y1) * ia; r[7] = (x1 - x0) * ia; r[8] = (x0 * y1 - y0 * x1) * ia;
  r[9] = z0; r[10] = z1; r[11] = z2;
  r[12] = z1 * z2; r[13] = z0 * z2; r[14] = z0 * z1;
  r[15] = ok ? 1.0f : 0.0f;
}

// --------------------------- kernel 3: WMMA rasterizer ---------------------
__global__ __launch_bounds__(RASTER_BLOCK) void raster(
    const float* __restrict__ frec,
    const int* __restrict__ pH, const int* __restrict__ pW,
    float* __restrict__ outFace, float* __restrict__ outZ,
    float* __restrict__ outBary, float* __restrict__ outDist,
    float* __restrict__ pixRec, int Fn, int HW) {
  __shared__ float4 lds4[CH * 4];   // CH faces * 16 floats, 16B aligned
  float* lds = (float*)lds4;

  const int bat = blockIdx.y;
  const int tid = threadIdx.x;
  const int wave = tid >> 5;
  const int lane = tid & 31;
  const int lane16 = lane & 15;
  const bool lo = lane < 16;
  const int H = *pH, W = *pW;
  const float invW = 1.0f / (float)W, invH = 1.0f / (float)H;

  const int strip = blockIdx.x * RASTER_WAVES + wave;
  const int pbase = strip * 16;

  // A operand (constant over the face loop):
  // lanes 0-15 : {K0=px, K1=py} for pixel row M=lane
  // lanes 16-31: {K2=1,  K3=0 }
  v2f a;
  {
    int p = pbase + lane16; if (p >= HW) p = HW - 1;
    const int iy = p / W, ix = p - iy * W;
    const float px = 1.0f - (2.0f * (float)ix + 1.0f) * invW;
    const float py = 1.0f - (2.0f * (float)iy + 1.0f) * invH;
    a.x = lo ? px : 1.0f;
    a.y = lo ? py : 0.0f;
  }

  float zbest[8];
  int   ibest[8];
#pragma unroll
  for (int r = 0; r < 8; ++r) { zbest[r] = __builtin_inff(); ibest[r] = 0x7fffffff; }

  const float* fb = frec + (size_t)bat * Fn * 16;
  const int nchunks = (Fn + CH - 1) / CH;

  for (int c = 0; c < nchunks; ++c) {
    __syncthreads();
    const int fbase = c * CH;
    // stage CH face records into LDS (float4 granularity, 2 per thread)
    for (int i = tid; i < CH * 4; i += RASTER_BLOCK) {
      const int f = fbase + (i >> 2);
      const int fc = f < Fn ? f : Fn - 1;
      float4 val = ((const float4*)(fb + (size_t)fc * 16))[i & 3];
      if (f >= Fn && (i & 3) == 3) val.w = 0.0f;  // kill ok-flag on padding
      lds4[i] = val;
    }
    if (c + 1 < nchunks) {  // global_prefetch_b8 of next chunk
      __builtin_prefetch(fb + (size_t)(c + 1) * CH * 16 + tid * 8, 0, 0);
    }
    __syncthreads();

#pragma unroll 1
    for (int s = 0; s < CH / 16; ++s) {
      const float* fl = lds + s * 16 * 16;       // this step's 16 faces
      const float* fr = fl + lane16 * 16;        // this lane's face column
      // B operands: low half supplies rows K0(A_e),K1(B_e); high half K2(C_e),K3(0)
      v2f b0, b1, b2;
      b0.x = lo ? fr[0] : fr[2];  b0.y = lo ? fr[1] : 0.0f;
      b1.x = lo ? fr[3] : fr[5];  b1.y = lo ? fr[4] : 0.0f;
      b2.x = lo ? fr[6] : fr[8];  b2.y = lo ? fr[7] : 0.0f;
      const v8f zero = {};
      v8f w0a = __builtin_amdgcn_wmma_f32_16x16x4_f32(false, a, false, b0, (short)0, zero, false, false);
      v8f w1a = __builtin_amdgcn_wmma_f32_16x16x4_f32(false, a, false, b1, (short)0, zero, false, false);
      v8f w2a = __builtin_amdgcn_wmma_f32_16x16x4_f32(false, a, false, b2, (short)0, zero, false, false);

      const float z0 = fr[9],  z1 = fr[10], z2 = fr[11];
      const float zz0 = fr[12], zz1 = fr[13], zz2 = fr[14];
      const float okf = fr[15];
      const int fg = fbase + s * 16 + lane16;
#pragma unroll
      for (int r = 0; r < 8; ++r) {
        const float w0 = w0a[r], w1 = w1a[r], w2 = w2a[r];
        const float l0 = w0 * zz0, l1 = w1 * zz1, l2 = w2 * zz2;
        const float ssum = l0 + l1 + l2;
        const float sA = (ssum == 0.0f) ? 1.0f : ssum;
        const float inv = __builtin_amdgcn_rcpf(sA);
        const float b0v = fmaxf(l0 * inv, 0.0f);
        const float b1v = fmaxf(l1 * inv, 0.0f);
        const float b2v = fmaxf(l2 * inv, 0.0f);
        const float bs = fmaxf(b0v + b1v + b2v, RAST_EPS);
        const float zp = (b0v * z0 + b1v * z1 + b2v * z2) * __builtin_amdgcn_rcpf(bs);
        const bool inside = (w0 >= 0.0f) && (w1 >= 0.0f) && (w2 >= 0.0f) && (okf > 0.5f);
        const float zm = inside ? zp : __builtin_inff();
        const bool better = zm < zbest[r];   // strict: first face wins ties
        zbest[r] = better ? zm : zbest[r];
        ibest[r] = better ? fg : ibest[r];
      }
    }
  }

  // cross-face reduction: butterfly over the 16 lanes of each half
#pragma unroll
  for (int r = 0; r < 8; ++r) {
    float z = zbest[r]; int i = ibest[r];
#pragma unroll
    for (int m = 1; m < 16; m <<= 1) {
      const float oz = __shfl_xor(z, m, 32);
      const int   oi = __shfl_xor(i, m, 32);
      if (oz < z || (oz == z && oi < i)) { z = oz; i = oi; }
    }
    zbest[r] = z; ibest[r] = i;
  }

  if (lane == 0 || lane == 16) {
    const int half = lane >> 4;
    for (int r = 0; r < 8; ++r) {
      const int p = pbase + half * 8 + r;
      if (p >= HW) continue;
      const float z = zbest[r];
      const int fid = ibest[r];
      const bool hit = z < 3.0e38f;  // isfinite(zmin)
      float b0v = -1.0f, b1v = -1.0f, b2v = -1.0f;
      if (hit) {  // recompute bary of the winning face
        const float* fr = fb + (size_t)fid * 16;
        const int iy = p / W, ix = p - iy * W;
        const float px = 1.0f - (2.0f * (float)ix + 1.0f) * invW;
        const float py = 1.0f - (2.0f * (float)iy + 1.0f) * invH;
        const float w0 = fr[0] * px + fr[1] * py + fr[2];
        const float w1 = fr[3] * px + fr[4] * py + fr[5];
        const float w2 = fr[6] * px + fr[7] * py + fr[8];
        const float l0 = w0 * fr[12], l1 = w1 * fr[13], l2 = w2 * fr[14];
        const float ssum = l0 + l1 + l2;
        const float sA = (ssum == 0.0f) ? 1.0f : ssum;
        b0v = fmaxf(l0 / sA, 0.0f);
        b1v = fmaxf(l1 / sA, 0.0f);
        b2v = fmaxf(l2 / sA, 0.0f);
        const float bs = fmaxf(b0v + b1v + b2v, RAST_EPS);
        b0v /= bs; b1v /= bs; b2v /= bs;
      }
      const size_t g = (size_t)bat * HW + p;
      outFace[g] = hit ? (float)(fid + bat * Fn) : -1.0f;
      outZ[g]    = hit ? z : -1.0f;
      outBary[g * 3 + 0] = b0v;
      outBary[g * 3 + 1] = b1v;
      outBary[g * 3 + 2] = b2v;
      outDist[g] = hit ? 0.0f : -1.0f;
      float* rec = pixRec + g * 8;
      rec[0] = __int_as_float(fid);
      rec[1] = b0v; rec[2] = b1v; rec[3] = b2v;
      rec[4] = hit ? 1.0f : 0.0f;
      rec[5] = z; rec[6] = 0.0f; rec[7] = 0.0f;
    }
  }
}

// --------------------------- kernel 4: feature shading ---------------------
__global__ void shade(const float* __restrict__ pixRec,
                      const int* __restrict__ faces,
                      const float* __restrict__ feats,
                      float* __restrict__ outFeat,
                      int V, int Fn, int C, int HW) {
  const int b = blockIdx.y;
  const size_t t = (size_t)blockIdx.x * blockDim.x + threadIdx.x;
  const size_t total = (size_t)HW * C;
  if (t >= total) return;
  const int p = (int)(t / C);
  const int c = (int)(t - (size_t)p * C);
  const float* rec = pixRec + ((size_t)b * HW + p) * 8;
  float out = 0.0f;
  if (rec[4] > 0.5f) {
    const int fid = __float_as_int(rec[0]);
    const int* fv = faces + ((size_t)b * Fn + fid) * 3;
    const float* fbt = feats + (size_t)b * V * C;
    out = rec[1] * fbt[(size_t)fv[0] * C + c]
        + rec[2] * fbt[(size_t)fv[1] * C + c]
        + rec[3] * fbt[(size_t)fv[2] * C + c];
  }
  outFeat[((size_t)b * HW + p) * C + c] = out;
}

// --------------------------- host entry ------------------------------------
extern "C" void kernel_launch(void* const* d_in, const int* in_sizes, int n_in,
                              void* d_out, int out_size, void* d_ws, size_t ws_size,
                              hipStream_t stream) {
  const float* positions = (const float*)d_in[0];
  const float* features  = (const float*)d_in[1];
  const int*   faces     = (const int*)d_in[2];
  const float* Kmat      = (const float*)d_in[3];
  const float* RT        = (const float*)d_in[4];
  const int*   pH        = (const int*)d_in[5];
  const int*   pW        = (const int*)d_in[6];
  // d_in[7] = blur_sigma (unused, as in the reference)

  const int B  = in_sizes[3] / 9;
  const int V  = in_sizes[0] / (3 * B);
  const int C  = in_sizes[1] / (B * V);
  const int Fn = in_sizes[2] / (3 * B);
  const int HW = out_size / (B * (C + 6));   // out = B*H*W*(C+1+1+3+1)

  float* wsf    = (float*)d_ws;
  float* verts  = wsf;                                // B*V*4
  float* frec   = verts + (size_t)B * V * 4;          // B*Fn*16
  float* pixRec = frec + (size_t)B * Fn * 16;         // B*HW*8

  float* out0 = (float*)d_out;                        // pix_feats  B*HW*C
  float* out1 = out0 + (size_t)B * HW * C;            // pix_to_face B*HW
  float* out2 = out1 + (size_t)B * HW;                // zbuf        B*HW
  float* out3 = out2 + (size_t)B * HW;                // bary        B*HW*3
  float* out4 = out3 + (size_t)B * HW * 3;            // dists       B*HW

  dim3 gv((V + 255) / 256, B);
  vertex_xform<<<gv, 256, 0, stream>>>(positions, Kmat, RT, pH, pW, verts, V);

  dim3 gf((Fn + 255) / 256, B);
  face_setup<<<gf, 256, 0, stream>>>(faces, verts, frec, V, Fn);

  const int strips = (HW + 15) / 16;
  dim3 gr((strips + RASTER_WAVES - 1) / RASTER_WAVES, B);
  raster<<<gr, RASTER_BLOCK, 0, stream>>>(frec, pH, pW, out1, out2, out3, out4,
                                          pixRec, Fn, HW);

  const size_t tot = (size_t)HW * C;
  dim3 gs((unsigned)((tot + 255) / 256), B);
  shade<<<gs, 256, 0, stream>>>(pixRec, faces, features, out0, V, Fn, C, HW);
}